// Conv2D_33079838114353
// MI455X (gfx1250) — compile-verified
//
#include <hip/hip_runtime.h>

typedef __attribute__((ext_vector_type(2))) float v2f;
typedef __attribute__((ext_vector_type(8))) float v8f;

namespace {
constexpr int H = 8192, W = 8192;
constexpr int OH = H - 2, OW = W - 2;          // 8190 x 8190 valid conv output
constexpr int TILE_Y = 32, TILE_X = 64;        // per-block output region
constexpr int LDS_H = TILE_Y + 2;              // 34
constexpr int LDS_W = TILE_X + 2;              // 66
constexpr int LDS_STRIDE = 68;                 // padded row stride (floats)
constexpr int KDIM = 56;                       // 3*18 = 54, padded to mult of 4
}

// D = A x B + C with V_WMMA_F32_16X16X4_F32 (fp32 matrix pipe, wave32).
// k = 18*i + c ; A[m][k] = in[y0+m+i][x0+c] ; B[k][n] = w[i][c-n] (banded).
__global__ __launch_bounds__(256)
void conv3x3_wmma_f32(const float* __restrict__ x,
                      const float* __restrict__ wgt,
                      const float* __restrict__ bias,
                      float* __restrict__ out) {
  __shared__ float tile[LDS_H * LDS_STRIDE];   // 34x66 input window (stride 68)
  __shared__ float bmat[KDIM * 16];            // banded weight matrix B[k][n]

  const int tid = threadIdx.x;
  const int oy0 = blockIdx.y * TILE_Y;
  const int ox0 = blockIdx.x * TILE_X;

  // ---- Stage input window into LDS (coalesced, zero-fill out-of-range) ----
  for (int idx = tid; idx < LDS_H * LDS_W; idx += 256) {
    int r = idx / LDS_W;
    int c = idx - r * LDS_W;
    int gr = oy0 + r, gc = ox0 + c;
    float v = 0.0f;
    if (gr < H && gc < W) v = x[gr * W + gc];
    tile[r * LDS_STRIDE + c] = v;
  }

  // ---- Build banded weight matrix once per block ----
  for (int idx = tid; idx < KDIM * 16; idx += 256) {
    int k = idx >> 4;
    int n = idx & 15;
    float v = 0.0f;
    if (k < 54) {
      int i = (k >= 36) ? 2 : ((k >= 18) ? 1 : 0);
      int c = k - 18 * i;
      int j = c - n;
      if (j >= 0 && j < 3) v = wgt[i * 3 + j];
    }
    bmat[idx] = v;
  }
  __syncthreads();

  // ---- Per-wave 16x16 output tile via 14 chained f32 WMMAs ----
  const int lane = tid & 31;
  const int wave = tid >> 5;          // 0..7
  const int wty  = wave >> 2;         // 0..1  (tile row within block)
  const int wtx  = wave & 3;          // 0..3  (tile col within block)
  const int mn   = lane & 15;         // A row (M) == B col (N) index
  const int koff = (lane >> 4) << 1;  // lanes 0-15 -> K+0/K+1, 16-31 -> K+2/K+3

  const int arow = wty * 16 + mn;     // base LDS row for this lane's A row
  const int acol = wtx * 16;          // base LDS col for this tile

  v8f acc = {0.f, 0.f, 0.f, 0.f, 0.f, 0.f, 0.f, 0.f};

#pragma unroll
  for (int f = 0; f < 14; ++f) {
    const int k0 = 4 * f + koff;
    float a0, a1;
    {
      int k = k0;
      int i = (k >= 36) ? 2 : ((k >= 18) ? 1 : 0);
      int c = k - 18 * i;
      a0 = (k < 54) ? tile[(arow + i) * LDS_STRIDE + acol + c] : 0.0f;
    }
    {
      int k = k0 + 1;
      int i = (k >= 36) ? 2 : ((k >= 18) ? 1 : 0);
      int c = k - 18 * i;
      a1 = (k < 54) ? tile[(arow + i) * LDS_STRIDE + acol + c] : 0.0f;
    }
    v2f a; a.x = a0;                    a.y = a1;
    v2f b; b.x = bmat[k0 * 16 + mn];    b.y = bmat[(k0 + 1) * 16 + mn];
    // 8 args: (neg_a, A, neg_b, B, c_mod, C, reuse_a, reuse_b)
    acc = __builtin_amdgcn_wmma_f32_16x16x4_f32(false, a, false, b,
                                                (short)0, acc, false, false);
  }

  // ---- Store 16x16 tile: VGPR v holds M=v (lanes 0-15) / M=v+8 (16-31) ----
  const float bv = bias[0];
  const int orow0 = oy0 + wty * 16 + ((lane >> 4) << 3);
  const int ocol  = ox0 + wtx * 16 + mn;
#pragma unroll
  for (int v = 0; v < 8; ++v) {
    int orow = orow0 + v;
    if (orow < OH && ocol < OW)
      out[orow * OW + ocol] = acc[v] + bv;
  }
}

extern "C" void kernel_launch(void* const* d_in, const int* in_sizes, int n_in,
                              void* d_out, int out_size, void* d_ws, size_t ws_size,
                              hipStream_t stream) {
  const float* x    = (const float*)d_in[0];   // [8192, 8192] fp32
  const float* wgt  = (const float*)d_in[1];   // [3, 3] fp32
  const float* bias = (const float*)d_in[2];   // [1] fp32
  float* out = (float*)d_out;                  // [8190, 8190] fp32

  dim3 grid((OW + TILE_X - 1) / TILE_X,        // 128
            (OH + TILE_Y - 1) / TILE_Y);       // 256
  conv3x3_wmma_f32<<<grid, dim3(256), 0, stream>>>(x, wgt, bias, out);
}